// SEBlock_49340584296670
// MI455X (gfx1250) — compile-verified
//
#include <hip/hip_runtime.h>
#include <hip/hip_bf16.h>
#include <math.h>

typedef __attribute__((ext_vector_type(2))) float v2f;
typedef __attribute__((ext_vector_type(8))) float v8f;

#define HW   3136      // 56*56
#define HW4  784       // HW / 4
#define CCH  512
#define MID  32
#define BB   32

// ---------------------------------------------------------------------------
// Kernel 1: squeeze — per-(b,c) mean over 3136 spatial elements.
// One block per (b,c). 128 threads, float4 (b128) coalesced reads.
// ---------------------------------------------------------------------------
__global__ void __launch_bounds__(128) se_squeeze(const float* __restrict__ x,
                                                  float* __restrict__ s) {
    const int bc = blockIdx.x;
    const float4* xv = (const float4*)(x + (size_t)bc * HW);

    float sum = 0.0f;
    for (int i = threadIdx.x; i < HW4; i += 128) {
        float4 v = xv[i];
        sum += (v.x + v.y) + (v.z + v.w);
    }

    __shared__ float red[128];
    red[threadIdx.x] = sum;
    __syncthreads();
    for (int off = 64; off > 0; off >>= 1) {
        if (threadIdx.x < off) red[threadIdx.x] += red[threadIdx.x + off];
        __syncthreads();
    }
    if (threadIdx.x == 0) s[bc] = red[0] * (1.0f / (float)HW);
}

// ---------------------------------------------------------------------------
// Kernel 2: excitation — two small GEMMs on the WMMA f32 16x16x4 path.
// Single wave (32 threads), EXEC all 1s as WMMA requires.
//
// Fragment layout (V_WMMA_F32_16X16X4_F32, wave32):
//   A (16x4): lane L holds row (L&15); K = {0,1} for lanes 0-15, {2,3} for 16-31
//   B (4x16): lane L holds col (L&15); same K split              -> float2/lane
//   C/D (16x16): VGPR v -> row v + 8*(L>=16), col (L&15)         -> v8f/lane
// ---------------------------------------------------------------------------
__global__ void __launch_bounds__(32) se_excite(const float* __restrict__ s,   // [32][512]
                                                const float* __restrict__ w1,  // [32][512]
                                                const float* __restrict__ b1,  // [32]
                                                const float* __restrict__ w2,  // [512][32]
                                                const float* __restrict__ b2,  // [512]
                                                float* __restrict__ g) {       // [32][512]
    __shared__ float h[BB][MID];   // relu(s @ w1^T + b1)

    const int lane   = threadIdx.x;
    const int r      = lane & 15;          // row/col within 16-wide tile
    const int khalf  = (lane >> 4) << 1;   // 0 or 2 (K sub-offset per half-wave)
    const int rowAdd = (lane >> 4) << 3;   // 0 or 8 (C/D row offset per half-wave)

    // ---- GEMM1: h[b][m] = relu( sum_c s[b][c] * w1[m][c] + b1[m] ) ----
    // M = 32 (b), N = 32 (m), K = 512 (c): 2x2 tiles, 128 K-steps of 4.
    v8f acc00 = {}, acc01 = {}, acc10 = {}, acc11 = {};
    for (int k = 0; k < CCH; k += 4) {
        const int kb = k + khalf;
        v2f a0 = *(const v2f*)(s  + (size_t)(r +  0) * CCH + kb);   // A, M-tile 0
        v2f a1 = *(const v2f*)(s  + (size_t)(r + 16) * CCH + kb);   // A, M-tile 1
        v2f bf0 = *(const v2f*)(w1 + (size_t)(r +  0) * CCH + kb);  // B, N-tile 0
        v2f bf1 = *(const v2f*)(w1 + (size_t)(r + 16) * CCH + kb);  // B, N-tile 1
        acc00 = __builtin_amdgcn_wmma_f32_16x16x4_f32(false, a0, false, bf0, (short)0, acc00, false, false);
        acc01 = __builtin_amdgcn_wmma_f32_16x16x4_f32(false, a0, false, bf1, (short)0, acc01, false, false);
        acc10 = __builtin_amdgcn_wmma_f32_16x16x4_f32(false, a1, false, bf0, (short)0, acc10, false, false);
        acc11 = __builtin_amdgcn_wmma_f32_16x16x4_f32(false, a1, false, bf1, (short)0, acc11, false, false);
    }

    // Bias + ReLU, write h to LDS.
    #pragma unroll
    for (int v = 0; v < 8; ++v) {
        h[ 0 + v + rowAdd][ 0 + r] = fmaxf(acc00[v] + b1[ 0 + r], 0.0f);
        h[ 0 + v + rowAdd][16 + r] = fmaxf(acc01[v] + b1[16 + r], 0.0f);
        h[16 + v + rowAdd][ 0 + r] = fmaxf(acc10[v] + b1[ 0 + r], 0.0f);
        h[16 + v + rowAdd][16 + r] = fmaxf(acc11[v] + b1[16 + r], 0.0f);
    }
    __syncthreads();

    // ---- GEMM2: g[b][c] = sigmoid( sum_m h[b][m] * w2[c][m] + b2[c] ) ----
    // M = 32 (b), N = 512 (c), K = 32 (m): 32 N-tiles x 2 M-tiles, 8 K-steps.
    for (int nt = 0; nt < CCH / 16; ++nt) {
        const int nbase = nt * 16;
        v8f acc0 = {}, acc1 = {};
        #pragma unroll
        for (int k = 0; k < MID; k += 4) {
            const int kb = k + khalf;
            v2f af0 = *(const v2f*)(&h[ 0 + r][kb]);                          // A from LDS, M-tile 0
            v2f af1 = *(const v2f*)(&h[16 + r][kb]);                          // A from LDS, M-tile 1
            v2f bf  = *(const v2f*)(w2 + (size_t)(nbase + r) * MID + kb);     // B, shared by both M-tiles
            acc0 = __builtin_amdgcn_wmma_f32_16x16x4_f32(false, af0, false, bf, (short)0, acc0, false, false);
            acc1 = __builtin_amdgcn_wmma_f32_16x16x4_f32(false, af1, false, bf, (short)0, acc1, false, false);
        }
        const float bias = b2[nbase + r];
        #pragma unroll
        for (int v = 0; v < 8; ++v) {
            float t0 = acc0[v] + bias;
            float t1 = acc1[v] + bias;
            g[(size_t)( 0 + v + rowAdd) * CCH + nbase + r] = 1.0f / (1.0f + __expf(-t0));
            g[(size_t)(16 + v + rowAdd) * CCH + nbase + r] = 1.0f / (1.0f + __expf(-t1));
        }
    }
}

// ---------------------------------------------------------------------------
// Kernel 3: scale — out = x * g[b][c], float4 coalesced read+write.
// ---------------------------------------------------------------------------
__global__ void __launch_bounds__(256) se_scale(const float* __restrict__ x,
                                                const float* __restrict__ g,
                                                float* __restrict__ out) {
    const unsigned idx = blockIdx.x * 256u + threadIdx.x;  // float4 index
    const unsigned bc  = idx / HW4;                         // (b*C + c); g is tiny & L2-hot
    const float gv = g[bc];
    float4 v = ((const float4*)x)[idx];
    v.x *= gv; v.y *= gv; v.z *= gv; v.w *= gv;
    ((float4*)out)[idx] = v;
}

extern "C" void kernel_launch(void* const* d_in, const int* in_sizes, int n_in,
                              void* d_out, int out_size, void* d_ws, size_t ws_size,
                              hipStream_t stream) {
    const float* x  = (const float*)d_in[0];
    const float* w1 = (const float*)d_in[1];
    const float* b1 = (const float*)d_in[2];
    const float* w2 = (const float*)d_in[3];
    const float* b2 = (const float*)d_in[4];
    float* out = (float*)d_out;

    float* s = (float*)d_ws;           // [32][512]  = 64 KB
    float* g = s + BB * CCH;           // [32][512]  = 64 KB

    // Pass 1: read x once (205 MB) -> per-channel means
    se_squeeze<<<dim3(BB * CCH), dim3(128), 0, stream>>>(x, s);
    // Tiny excitation GEMMs on the WMMA f32 pipe (single wave)
    se_excite<<<dim3(1), dim3(32), 0, stream>>>(s, w1, b1, w2, b2, g);
    // Pass 2: read x + write out (410 MB)
    const unsigned nvec4 = (unsigned)BB * CCH * HW4;   // 12,845,056
    se_scale<<<dim3(nvec4 / 256), dim3(256), 0, stream>>>(x, g, out);
}